// CrossNet_5609227289106
// MI455X (gfx1250) — compile-verified
//
#include <hip/hip_runtime.h>
#include <hip/hip_bf16.h>

// CrossNet collapse:  x_l = alpha_l * x0 + c_l   (c_l = prefix-sum of biases, row-independent)
//   g_l  = dot(x0_row, w_l)        -> GEMM  G = X0 * W^T  (v_wmma_f32_16x16x4_f32, full f32)
//   t_l  = dot(c_l, w_l)           -> tiny precompute
//   s_l  = alpha_l * g_l + t_l ;  alpha_{l+1} = alpha_l + s_l ; alpha_0 = 1
//   out  = alpha_L * x0 + sum_l b_l
//
// B=16384, D=1024, L=4. Memory-bound: ~128MB min HBM traffic (~5.5us @ 23.3 TB/s).
// X0 is streamed once HBM->LDS via global_load_async_to_lds_b128 (ASYNCcnt),
// re-read in the epilogue from the 192MB L2. Gram stage runs on the WMMA pipe.

#define NB   16384
#define ND   1024
#define NL   4

typedef float v2f __attribute__((ext_vector_type(2)));
typedef float v8f __attribute__((ext_vector_type(8)));
typedef int   v4i __attribute__((ext_vector_type(4)));

// ---- CDNA5 async global->LDS path (guarded probe, safe fallback) ----------
#if defined(__AMDGCN__) && __has_builtin(__builtin_amdgcn_global_load_async_to_lds_b128)
#define HAVE_ASYNC_LDS 1
typedef __attribute__((address_space(1))) v4i* g_v4i_ptr;
typedef __attribute__((address_space(3))) v4i* l_v4i_ptr;
#else
#define HAVE_ASYNC_LDS 0
#endif

__device__ __forceinline__ void copy16_to_lds(const float* __restrict__ gsrc,
                                              float* __restrict__ ldst) {
#if HAVE_ASYNC_LDS
    __builtin_amdgcn_global_load_async_to_lds_b128(
        (g_v4i_ptr)gsrc, (l_v4i_ptr)ldst, /*imm offset=*/0, /*cpol=*/0);
#else
    *(float4*)ldst = *(const float4*)gsrc;
#endif
}

__device__ __forceinline__ void async_wait0() {
#if HAVE_ASYNC_LDS
#if __has_builtin(__builtin_amdgcn_s_wait_asynccnt)
    __builtin_amdgcn_s_wait_asynccnt(0);
#else
    asm volatile("s_wait_asynccnt 0" ::: "memory");
#endif
#endif
}

// ---------------------------------------------------------------------------
// Kernel 1: t_l = dot(prefix-bias_l, w_l)  and  csum[d] = sum_l bias[l][d]
// ---------------------------------------------------------------------------
__global__ __launch_bounds__(256)
void crossnet_precompute(const float* __restrict__ w,
                         const float* __restrict__ bias,
                         float* __restrict__ T,      // [4]
                         float* __restrict__ csum)   // [1024]
{
    __shared__ float red[3][256];
    const int tid = threadIdx.x;
    float t1 = 0.f, t2 = 0.f, t3 = 0.f;
    #pragma unroll
    for (int ii = 0; ii < ND / 256; ++ii) {
        const int d = ii * 256 + tid;
        const float b0 = bias[0 * ND + d];
        const float b1 = bias[1 * ND + d];
        const float b2 = bias[2 * ND + d];
        const float b3 = bias[3 * ND + d];
        csum[d] = b0 + b1 + b2 + b3;
        const float c1 = b0;            t1 += c1 * w[1 * ND + d];
        const float c2 = c1 + b1;       t2 += c2 * w[2 * ND + d];
        const float c3 = c2 + b2;       t3 += c3 * w[3 * ND + d];
    }
    red[0][tid] = t1; red[1][tid] = t2; red[2][tid] = t3;
    __syncthreads();
    for (int s = 128; s > 0; s >>= 1) {
        if (tid < s) {
            red[0][tid] += red[0][tid + s];
            red[1][tid] += red[1][tid + s];
            red[2][tid] += red[2][tid + s];
        }
        __syncthreads();
    }
    if (tid == 0) {
        T[0] = 0.f;
        T[1] = red[0][0];
        T[2] = red[1][0];
        T[3] = red[2][0];
    }
}

// ---------------------------------------------------------------------------
// Kernel 2: fused  G = X0*W^T (WMMA f32)  ->  alpha recurrence  ->  output
// Block = 256 threads (8 wave32), owns 128 rows (16 rows per wave).
// ---------------------------------------------------------------------------
#define SLAB_K    64          // K-columns staged per slab
#define SLAB_PAD  68          // LDS row stride in floats (bank-conflict pad, 16B aligned)

__global__ __launch_bounds__(256)
void crossnet_main(const float* __restrict__ x0,
                   const float* __restrict__ w,
                   const float* __restrict__ T,
                   const float* __restrict__ csum,
                   float* __restrict__ out)
{
    __shared__ float wlds[NL * ND];                 // 16 KB: all weights
    __shared__ float slab[128 * SLAB_PAD];          // 34 KB: X0 slab, 128 rows x 64 cols
    __shared__ float Gs[8][16][NL];                 // per-wave Gram columns
    __shared__ float alphas[128];

    const int  tid   = threadIdx.x;
    const int  wave  = tid >> 5;       // 0..7
    const int  lane  = tid & 31;       // wave32
    const int  nIdx  = lane & 15;      // N column (layer index when < 4)
    const int  kHalf = lane >> 4;      // 0: K {0,1}   1: K {2,3}   (f32 A/B layout)
    const bool bValid = (nIdx < NL);   // hoisted: one v_cmp, then v_cndmask per step
    const int  bRow  = bValid ? nIdx : 0;   // clamped B row (garbage cols are discarded)
    const long tileRow = (long)blockIdx.x * 128;

    // stage all weights in LDS (async path; 1024 float4 / 256 threads = 4 each)
    #pragma unroll
    for (int ii = 0; ii < (NL * ND) / (256 * 4); ++ii) {
        const int i4 = ii * 256 + tid;
        copy16_to_lds(w + i4 * 4, &wlds[i4 * 4]);
    }
    async_wait0();
    __syncthreads();

    v8f acc = {0.f, 0.f, 0.f, 0.f, 0.f, 0.f, 0.f, 0.f};
    const int myRow = wave * 16 + nIdx;            // A-frag row within 128-row tile

    for (int k0 = 0; k0 < ND; k0 += SLAB_K) {
        // stage slab: 128 rows x 64 cols = 2048 float4, 8 per thread, coalesced
        #pragma unroll
        for (int ii = 0; ii < 8; ++ii) {
            const int i  = ii * 256 + tid;
            const int r  = i >> 4;                 // row 0..127
            const int c4 = i & 15;                 // float4 column
            copy16_to_lds(x0 + (tileRow + r) * ND + k0 + c4 * 4,
                          &slab[r * SLAB_PAD + c4 * 4]);
        }
        async_wait0();
        __syncthreads();

        // 16 WMMA steps (K += 4 each) over this slab
        #pragma unroll
        for (int k = 0; k < SLAB_K; k += 4) {
            // A: 16x4 f32 tile of X0.  lane<16 -> K {k,k+1}; lane>=16 -> K {k+2,k+3}
            const v2f a = *(const v2f*)&slab[myRow * SLAB_PAD + k + 2 * kHalf];
            // B: 4x16 f32 tile of W^T; only N = 0..3 columns carry real weights.
            // Unconditional clamped load + select: WMMA D column n depends only on
            // B column n, so zero-selected lanes 4..15 keep the live columns exact.
            const v2f bw = *(const v2f*)&wlds[bRow * ND + k0 + k + 2 * kHalf];
            v2f b;
            b.x = bValid ? bw.x : 0.f;
            b.y = bValid ? bw.y : 0.f;
            acc = __builtin_amdgcn_wmma_f32_16x16x4_f32(
                /*neg_a=*/false, a, /*neg_b=*/false, b,
                /*c_mod=*/(short)0, acc, /*reuse_a=*/false, /*reuse_b=*/false);
        }
        __syncthreads();
    }

    // D tile layout: VGPR j -> (M=j, N=lane) lanes 0-15 ; (M=j+8, N=lane-16) lanes 16-31
    if (nIdx < NL) {
        const int mbase = kHalf * 8;
        #pragma unroll
        for (int j = 0; j < 8; ++j) Gs[wave][mbase + j][nIdx] = acc[j];
    }
    __syncthreads();

    // alpha recurrence, one thread per row
    if (tid < 128) {
        const float t0 = T[0], t1 = T[1], t2 = T[2], t3 = T[3];
        const int wv = tid >> 4, rw = tid & 15;
        float alpha = 1.f;
        alpha += alpha * Gs[wv][rw][0] + t0;
        alpha += alpha * Gs[wv][rw][1] + t1;
        alpha += alpha * Gs[wv][rw][2] + t2;
        alpha += alpha * Gs[wv][rw][3] + t3;
        alphas[tid] = alpha;
    }
    __syncthreads();

    // out[r][d] = alpha[r] * x0[r][d] + csum[d]   (x0 re-read hits L2)
    // Per iteration: row = r (uniform), this thread covers cols tid*4..tid*4+3.
    const float4 cs = *(const float4*)(csum + tid * 4);   // loop-invariant
    #pragma unroll 4
    for (int r = 0; r < 128; ++r) {
        const long base = (tileRow + r) * ND + tid * 4;
        const float4 xv = *(const float4*)(x0 + base);
        const float  a  = alphas[r];
        float4 o;
        o.x = fmaf(xv.x, a, cs.x);
        o.y = fmaf(xv.y, a, cs.y);
        o.z = fmaf(xv.z, a, cs.z);
        o.w = fmaf(xv.w, a, cs.w);
        *(float4*)(out + base) = o;
    }
}

// ---------------------------------------------------------------------------
extern "C" void kernel_launch(void* const* d_in, const int* in_sizes, int n_in,
                              void* d_out, int out_size, void* d_ws, size_t ws_size,
                              hipStream_t stream) {
    const float* x0   = (const float*)d_in[0];   // [B, D]
    const float* w    = (const float*)d_in[1];   // [L, D]
    const float* bias = (const float*)d_in[2];   // [L, D]
    float* out  = (float*)d_out;                 // [B, D]
    float* T    = (float*)d_ws;                  // 4 floats
    float* csum = T + 4;                         // 1024 floats

    crossnet_precompute<<<1, 256, 0, stream>>>(w, bias, T, csum);
    crossnet_main<<<NB / 128, 256, 0, stream>>>(x0, w, T, csum, out);
}